// OneHotVectorEncoding_17884243821254
// MI455X (gfx1250) — compile-verified
//
#include <hip/hip_runtime.h>
#include <hip/hip_bf16.h>

// One-hot encoding: out[i, x[i]] = 1.0f, else 0.0f.
// [8192 rows x 32000 classes] float32 => ~1.05 GB of stores, 0 FLOPs.
// Pure store-bandwidth kernel:
//   - every output byte written exactly once (no memset+scatter)
//   - unconditional 128-bit NT stores (one base VGPR + immediate offsets)
//   - single 4-byte overwrite of the hot element by the SAME lane that wrote
//     its zero chunk (same-wave stores are ordered with stores on CDNA5)

typedef __attribute__((ext_vector_type(4))) float v4f;

namespace {
constexpr int NUM_CLASS = 32000;
constexpr int CHUNKS    = NUM_CLASS / 4;     // 8000 float4 chunks per row
constexpr int THREADS   = 320;               // 10 wave32s; 8000/320 = 25 exactly
constexpr int ITERS     = CHUNKS / THREADS;  // 25 stores per thread
}

__global__ __launch_bounds__(THREADS) void
onehot_kernel(const int* __restrict__ x, float* __restrict__ out)
{
    const unsigned row = blockIdx.x;              // 0..8191
    const unsigned h   = (unsigned)x[row];        // hot class; uniform -> s_load
    const unsigned hc  = h >> 2;                  // float4 chunk holding the 1.0
    const unsigned hot_lane = hc % THREADS;       // lane that writes that chunk

    float* __restrict__ rowp = out + (size_t)row * (size_t)NUM_CLASS;
    float* __restrict__ p    = rowp + 4u * threadIdx.x;

    const v4f zero = {0.0f, 0.0f, 0.0f, 0.0f};

    // 25 unconditional b128 NT stores; constant stride 5120 B folds into
    // the 24-bit immediate offset field -> near-zero VALU in steady state.
#pragma unroll
    for (int i = 0; i < ITERS; ++i) {
        __builtin_nontemporal_store(zero, (v4f*)(p + (size_t)i * (4u * THREADS)));
    }

    // The lane that just wrote the hot chunk overwrites the hot element.
    // Same wave, same address -> stores retire in program order.
    if (threadIdx.x == hot_lane) {
        __builtin_nontemporal_store(1.0f, rowp + h);
    }
}

extern "C" void kernel_launch(void* const* d_in, const int* in_sizes, int n_in,
                              void* d_out, int out_size, void* d_ws, size_t ws_size,
                              hipStream_t stream)
{
    const int* x   = (const int*)d_in[0];
    float* out     = (float*)d_out;
    const int rows = in_sizes[0];                 // 8 * 1024 = 8192

    onehot_kernel<<<dim3(rows), dim3(THREADS), 0, stream>>>(x, out);
}